// TimeDecoder_81973745811835
// MI455X (gfx1250) — compile-verified
//
#include <hip/hip_runtime.h>
#include <math.h>

#define Bn   32
#define Tn   4096
#define Dn   512
#define Sn   256
#define SCALE_INV 10.0f

typedef __attribute__((ext_vector_type(2))) float v2f;
typedef __attribute__((ext_vector_type(8))) float v8f;

// partials per batch: 16 blocks/b * 8 waves/block
#define NP 128
#define T_PER_WAVE (Tn / NP)   // 32

// workspace layout (float offsets)
#define WS_H      0
#define WS_Q      (WS_H + Bn*Dn)
#define WS_ML     (WS_Q + Bn*Dn)
#define WS_ACC    (WS_ML + Bn*NP*2)
#define WS_POOLED (WS_ACC + Bn*NP*Dn)

// ---------------- h = silu(t_star * W1 + b1) ----------------
__global__ void k_h(const float* __restrict__ t_star,
                    const float* __restrict__ W1,
                    const float* __restrict__ b1,
                    float* __restrict__ ws) {
    int i = blockIdx.x * blockDim.x + threadIdx.x;   // 0..B*D-1
    int b = i / Dn, d = i % Dn;
    float x = t_star[b] * W1[d] + b1[d];
    ws[WS_H + i] = x / (1.0f + __expf(-x));
}

// ---------------- C[32,N] = A[32,K] @ Bm[K,N] + bias, fp32 WMMA ----------------
// grid.x = N/16 tiles, grid.y = 2 (M tiles), 32 threads (1 wave) per block.
// split==0: Cout[m*N+n] = val
// split==1: N==2*S; mu -> Cout[m*S+n], log_sigma -> Cout[B*S + m*S + (n-S)]
__global__ void k_gemm_wmma(const float* __restrict__ A,
                            const float* __restrict__ Bm,
                            const float* __restrict__ bias,
                            float* __restrict__ Cout,
                            int N, int K, int split) {
    int lane = threadIdx.x & 31;
    int h    = lane >> 4;        // lane half selects K sub-pair / M offset
    int ln   = lane & 15;
    int m0   = blockIdx.y * 16;
    int n    = blockIdx.x * 16 + ln;

    v8f c = {};
    for (int k0 = 0; k0 < K; k0 += 4) {
        int ka = k0 + 2 * h;
        v2f a, bb;
        a.x  = A[(m0 + ln) * K + ka];
        a.y  = A[(m0 + ln) * K + ka + 1];
        bb.x = Bm[(size_t)(ka    ) * N + n];
        bb.y = Bm[(size_t)(ka + 1) * N + n];
        c = __builtin_amdgcn_wmma_f32_16x16x4_f32(false, a, false, bb,
                                                  (short)0, c, false, false);
    }
    float bv = bias[n];
    #pragma unroll
    for (int v = 0; v < 8; ++v) {
        int m = m0 + v + 8 * h;
        float val = c[v] + bv;
        if (!split) {
            Cout[m * N + n] = val;
        } else {
            if (n < Sn) Cout[m * Sn + n] = val;
            else        Cout[Bn * Sn + m * Sn + (n - Sn)] = val;
        }
    }
}

// ---------------- fused scores + online softmax + weighted accumulate ----------------
// grid.x = B*16 blocks, 256 threads (8 waves). Each wave: 32 consecutive t's.
__global__ void k_attend(const float* __restrict__ context,
                         const float* __restrict__ t_star,
                         const float* __restrict__ t_context,
                         float* __restrict__ ws) {
    int blk      = blockIdx.x;
    int b        = blk >> 4;           // / 16
    int blk_in_b = blk & 15;
    int wave     = threadIdx.x >> 5;
    int lane     = threadIdx.x & 31;
    int p        = blk_in_b * 8 + wave;        // 0..127
    int t0       = p * T_PER_WAVE;

    // preload q for this lane's 16 d's: d = lane*4 + j + 128*i
    const float* q = ws + WS_Q + b * Dn;
    float4 q4[4];
    #pragma unroll
    for (int i = 0; i < 4; ++i)
        q4[i] = *(const float4*)(q + lane * 4 + i * 128);

    float ts = t_star[b];
    const float* ctx_b = context   + (size_t)b * Tn * Dn;
    const float* tc_b  = t_context + (size_t)b * Tn;

    float m = -INFINITY, l = 0.0f;
    float4 acc[4] = {};

    for (int tt = 0; tt < T_PER_WAVE; ++tt) {
        int t = t0 + tt;
        const float* row = ctx_b + (size_t)t * Dn + lane * 4;
        float4 x[4];
        #pragma unroll
        for (int i = 0; i < 4; ++i)
            x[i] = *(const float4*)(row + i * 128);
        if (tt + 1 < T_PER_WAVE)
            __builtin_prefetch(row + Dn, 0, 1);   // global_prefetch_b8 next row

        float dotp = 0.0f;
        #pragma unroll
        for (int i = 0; i < 4; ++i)
            dotp += x[i].x * q4[i].x + x[i].y * q4[i].y +
                    x[i].z * q4[i].z + x[i].w * q4[i].w;
        #pragma unroll
        for (int off = 16; off > 0; off >>= 1)
            dotp += __shfl_xor(dotp, off, 32);

        float s  = dotp - fabsf(ts - tc_b[t]) * SCALE_INV;
        float mn = fmaxf(m, s);
        float sc = __expf(m - mn);     // 0 on first iteration (m = -inf)
        float w  = __expf(s - mn);
        l = l * sc + w;
        #pragma unroll
        for (int i = 0; i < 4; ++i) {
            acc[i].x = acc[i].x * sc + w * x[i].x;
            acc[i].y = acc[i].y * sc + w * x[i].y;
            acc[i].z = acc[i].z * sc + w * x[i].z;
            acc[i].w = acc[i].w * sc + w * x[i].w;
        }
        m = mn;
    }

    float* accout = ws + WS_ACC + ((size_t)(b * NP + p)) * Dn;
    #pragma unroll
    for (int i = 0; i < 4; ++i)
        *(float4*)(accout + lane * 4 + i * 128) = acc[i];
    if (lane == 0) {
        float* ml = ws + WS_ML + (b * NP + p) * 2;
        ml[0] = m; ml[1] = l;
    }
}

// ---------------- combine partials -> pooled[b,d] ----------------
__global__ void k_combine(float* __restrict__ ws) {
    int b = blockIdx.x;
    int d = threadIdx.x;                 // 512 threads
    const float* ml = ws + WS_ML + b * NP * 2;
    float M = -INFINITY;
    for (int p = 0; p < NP; ++p) M = fmaxf(M, ml[2 * p]);
    const float* accb = ws + WS_ACC + (size_t)b * NP * Dn;
    float L = 0.0f, sum = 0.0f;
    for (int p = 0; p < NP; ++p) {
        float e = __expf(ml[2 * p] - M);
        L   += e * ml[2 * p + 1];
        sum += e * accb[(size_t)p * Dn + d];
    }
    ws[WS_POOLED + b * Dn + d] = sum / L;
}

extern "C" void kernel_launch(void* const* d_in, const int* in_sizes, int n_in,
                              void* d_out, int out_size, void* d_ws, size_t ws_size,
                              hipStream_t stream) {
    const float* context   = (const float*)d_in[0];
    const float* t_star    = (const float*)d_in[1];
    const float* t_context = (const float*)d_in[2];
    const float* W1        = (const float*)d_in[3];
    const float* b1        = (const float*)d_in[4];
    const float* W2        = (const float*)d_in[5];
    const float* b2        = (const float*)d_in[6];
    const float* Wout      = (const float*)d_in[7];
    const float* bout      = (const float*)d_in[8];
    float* out = (float*)d_out;
    float* ws  = (float*)d_ws;

    // 1) h = silu(t_star*W1 + b1)
    k_h<<<(Bn * Dn) / 256, 256, 0, stream>>>(t_star, W1, b1, ws);

    // 2) q = h @ W2 + b2   (fp32 WMMA)
    k_gemm_wmma<<<dim3(Dn / 16, 2), 32, 0, stream>>>(ws + WS_H, W2, b2,
                                                     ws + WS_Q, Dn, Dn, 0);

    // 3) fused scores + online softmax + weighted context accumulate (single context pass)
    k_attend<<<Bn * 16, 256, 0, stream>>>(context, t_star, t_context, ws);

    // 4) combine partials -> pooled
    k_combine<<<Bn, Dn, 0, stream>>>(ws);

    // 5) out = pooled @ Wout + bout, split into mu / log_sigma
    k_gemm_wmma<<<dim3((2 * Sn) / 16, 2), 32, 0, stream>>>(ws + WS_POOLED, Wout, bout,
                                                           out, 2 * Sn, Dn, 1);
}